// SLICLayer_83648783057277
// MI455X (gfx1250) — compile-verified
//
#include <hip/hip_runtime.h>
#include <math.h>

typedef __attribute__((ext_vector_type(2))) float v2f;
typedef __attribute__((ext_vector_type(8))) float v8f;

#define BATCH 16
#define H 224
#define W 224
#define NPIX (H*W)         // 50176
#define K 100
#define KPAD 112           // 7 tiles of 16 centers
#define NT (NPIX/16)       // 3136 pixel tiles per image
#define CD 8               // padded feature dims (5 used + c2 slot + zeros)
#define COMPACT 10.0f

// ---------------------------------------------------------------------------
// Kernel 1: build padded features [b][pixel][8]:
//   {r/10, g/10, b/10, y/S, x/S, 0, 0, 0}
// ---------------------------------------------------------------------------
__global__ void k_feats(const float* __restrict__ img, float* __restrict__ feats) {
    int idx = blockIdx.x * blockDim.x + threadIdx.x;
    if (idx >= BATCH * NPIX) return;
    int b = idx / NPIX, p = idx % NPIX;
    int y = p / W, x = p % W;
    float S = sqrtf((float)(H * W) / (float)K);
    const float invC = 1.0f / COMPACT;
    const float* ib = img + (size_t)b * 3 * NPIX;
    float c0 = ib[0 * NPIX + p] * invC;
    float c1 = ib[1 * NPIX + p] * invC;
    float c2 = ib[2 * NPIX + p] * invC;
    float4* o = (float4*)(feats + (size_t)idx * CD);
    o[0] = make_float4(c0, c1, c2, (float)y / S);
    o[1] = make_float4((float)x / S, 0.0f, 0.0f, 0.0f);
}

// ---------------------------------------------------------------------------
// Kernel 2: seed centers from 10x10 grid pixels; slot 5 = |c|^2, pads get 1e30
// ---------------------------------------------------------------------------
__global__ void k_seed(const float* __restrict__ feats, float* __restrict__ centers) {
    int idx = blockIdx.x * blockDim.x + threadIdx.x;
    if (idx >= BATCH * KPAD) return;
    int b = idx / KPAD, c = idx % KPAD;
    float* o = centers + (size_t)idx * CD;
    if (c < K) {
        int sy = c / 10, sx = c % 10;
        int cy = (int)((sy + 0.5f) * ((float)H / 10.0f)); if (cy > H - 1) cy = H - 1;
        int cx = (int)((sx + 0.5f) * ((float)W / 10.0f)); if (cx > W - 1) cx = W - 1;
        const float* f = feats + ((size_t)b * NPIX + (size_t)cy * W + cx) * CD;
        float s2 = 0.0f;
        #pragma unroll
        for (int d = 0; d < 5; ++d) { float v = f[d]; o[d] = v; s2 += v * v; }
        o[5] = s2; o[6] = 0.0f; o[7] = 0.0f;
    } else {
        #pragma unroll
        for (int d = 0; d < 5; ++d) o[d] = 0.0f;
        o[5] = 1.0e30f; o[6] = 0.0f; o[7] = 0.0f;
    }
}

// ---------------------------------------------------------------------------
// Kernel 3: assignment via V_WMMA_F32_16X16X4_F32.
// One wave = one 16-pixel tile. A = 16x8 pixel feats (two 16x4 chunks),
// B = 4x16 center chunks, 7 center tiles -> full 16x112 cross matrix.
// dist = c2 - 2*cross (p2 constant per pixel, irrelevant for argmin).
// ---------------------------------------------------------------------------
__global__ void __launch_bounds__(256) k_assign(const float* __restrict__ feats,
                                                const float* __restrict__ centers,
                                                int* __restrict__ labels) {
    int wave = threadIdx.x >> 5;
    int lane = threadIdx.x & 31;
    int tileG = blockIdx.x * 8 + wave;       // grid sized exactly: no guard needed
    int b = tileG / NT;
    int t = tileG % NT;
    int baseP = t * 16;

    const float* fb = feats + ((size_t)b * NPIX + baseP) * CD;
    const float* cb = centers + (size_t)b * KPAD * CD;
    __builtin_prefetch(cb, 0, 3);

    int l16 = lane & 15;
    int hi  = lane >> 4;            // 0: K={0,1}/rows 0-7 ; 1: K={2,3}/rows 8-15
    int kA  = hi * 2;

    // A operands: pixel row M = l16, dims (kA,kA+1) and (4+kA,5+kA)
    v2f a0 = *(const v2f*)(fb + (size_t)l16 * CD + kA);
    v2f a1 = *(const v2f*)(fb + (size_t)l16 * CD + 4 + kA);

    float bestd[8];
    int   besti[8];
    #pragma unroll
    for (int r = 0; r < 8; ++r) { bestd[r] = 3.4e38f; besti[r] = 0; }

    #pragma unroll
    for (int t7 = 0; t7 < 7; ++t7) {
        int cidx = t7 * 16 + l16;                 // this lane's center column
        const float* cp = cb + (size_t)cidx * CD;
        v2f b0 = *(const v2f*)(cp + kA);          // B rows K=kA,kA+1 (col = l16)
        v2f b1 = *(const v2f*)(cp + 4 + kA);
        float c2 = cp[5];                         // |center|^2 (1e30 for pads)

        v8f acc = {};
        acc = __builtin_amdgcn_wmma_f32_16x16x4_f32(
                  false, a0, false, b0, (short)0, acc, false, false);
        acc = __builtin_amdgcn_wmma_f32_16x16x4_f32(
                  false, a1, false, b1, (short)0, acc, false, false);

        #pragma unroll
        for (int r = 0; r < 8; ++r) {
            float d = c2 - 2.0f * acc[r];
            if (d < bestd[r]) { bestd[r] = d; besti[r] = cidx; }
        }
    }

    // argmin across the 16 lanes of each half-wave (masks < 16 keep halves apart)
    #pragma unroll
    for (int m = 1; m < 16; m <<= 1) {
        #pragma unroll
        for (int r = 0; r < 8; ++r) {
            float od = __shfl_xor(bestd[r], m, 32);
            int   oi = __shfl_xor(besti[r], m, 32);
            if (od < bestd[r] || (od == bestd[r] && oi < besti[r])) {
                bestd[r] = od; besti[r] = oi;
            }
        }
    }

    if (l16 == 0) {
        int rowBase = baseP + hi * 8;             // lanes 0-15 -> rows 0-7, 16-31 -> 8-15
        #pragma unroll
        for (int r = 0; r < 8; ++r)
            labels[(size_t)b * NPIX + rowBase + r] = besti[r];
    }
}

// ---------------------------------------------------------------------------
// Kernel 4: zero segment sums
// ---------------------------------------------------------------------------
__global__ void k_zero(float* __restrict__ sums) {
    int idx = blockIdx.x * blockDim.x + threadIdx.x;
    if (idx < BATCH * K * 6) sums[idx] = 0.0f;
}

// ---------------------------------------------------------------------------
// Kernel 5: segment sums via LDS atomics (ds_add_f32) + global flush
// ---------------------------------------------------------------------------
__global__ void __launch_bounds__(256) k_accum(const float* __restrict__ feats,
                                               const int* __restrict__ labels,
                                               float* __restrict__ sums) {
    __shared__ float ls[K * 6];
    int b = blockIdx.x / 49;                 // NPIX/1024 == 49 chunks per image
    int chunk = blockIdx.x % 49;
    for (int i = threadIdx.x; i < K * 6; i += 256) ls[i] = 0.0f;
    __syncthreads();

    int base = chunk * 1024;
    #pragma unroll
    for (int j = 0; j < 4; ++j) {
        int p = base + j * 256 + threadIdx.x;
        const float* f = feats + ((size_t)b * NPIX + p) * CD;
        int lab = labels[(size_t)b * NPIX + p];
        float* s = &ls[lab * 6];
        atomicAdd(&s[0], f[0]);
        atomicAdd(&s[1], f[1]);
        atomicAdd(&s[2], f[2]);
        atomicAdd(&s[3], f[3]);
        atomicAdd(&s[4], f[4]);
        atomicAdd(&s[5], 1.0f);
    }
    __syncthreads();
    for (int i = threadIdx.x; i < K * 6; i += 256)
        atomicAdd(&sums[(size_t)b * K * 6 + i], ls[i]);
}

// ---------------------------------------------------------------------------
// Kernel 6: centers = sums/max(count,1); slot 5 = |c|^2; pads poisoned
// ---------------------------------------------------------------------------
__global__ void k_update(const float* __restrict__ sums, float* __restrict__ centers) {
    int idx = blockIdx.x * blockDim.x + threadIdx.x;
    if (idx >= BATCH * KPAD) return;
    int b = idx / KPAD, c = idx % KPAD;
    float* o = centers + (size_t)idx * CD;
    if (c < K) {
        const float* s = sums + ((size_t)b * K + c) * 6;
        float cnt = s[5]; if (cnt < 1.0f) cnt = 1.0f;
        float inv = 1.0f / cnt;
        float s2 = 0.0f;
        #pragma unroll
        for (int d = 0; d < 5; ++d) { float v = s[d] * inv; o[d] = v; s2 += v * v; }
        o[5] = s2; o[6] = 0.0f; o[7] = 0.0f;
    } else {
        #pragma unroll
        for (int d = 0; d < 5; ++d) o[d] = 0.0f;
        o[5] = 1.0e30f; o[6] = 0.0f; o[7] = 0.0f;
    }
}

// ---------------------------------------------------------------------------
extern "C" void kernel_launch(void* const* d_in, const int* in_sizes, int n_in,
                              void* d_out, int out_size, void* d_ws, size_t ws_size,
                              hipStream_t stream) {
    const float* img = (const float*)d_in[0];
    int* labels = (int*)d_out;               // int32 labels, B*H*W

    char* ws = (char*)d_ws;
    size_t featsBytes = (size_t)BATCH * NPIX * CD * sizeof(float);   // ~25.7 MB
    float* feats = (float*)ws;
    size_t off = (featsBytes + 255) & ~(size_t)255;
    float* centers = (float*)(ws + off);
    size_t centBytes = (size_t)BATCH * KPAD * CD * sizeof(float);
    off += (centBytes + 255) & ~(size_t)255;
    float* sums = (float*)(ws + off);

    k_feats<<<(BATCH * NPIX + 255) / 256, 256, 0, stream>>>(img, feats);
    k_seed<<<(BATCH * KPAD + 255) / 256, 256, 0, stream>>>(feats, centers);

    const int assignGrid = BATCH * NT / 8;   // 8 waves (tiles) per 256-thread block
    for (int it = 0; it < 10; ++it) {
        k_assign<<<assignGrid, 256, 0, stream>>>(feats, centers, labels);
        k_zero<<<(BATCH * K * 6 + 255) / 256, 256, 0, stream>>>(sums);
        k_accum<<<BATCH * 49, 256, 0, stream>>>(feats, labels, sums);
        k_update<<<(BATCH * KPAD + 255) / 256, 256, 0, stream>>>(sums, centers);
    }
    k_assign<<<assignGrid, 256, 0, stream>>>(feats, centers, labels);
}